// NonMaxSuppression_6983616823967
// MI455X (gfx1250) — compile-verified
//
#include <hip/hip_runtime.h>
#include <hip/hip_bf16.h>
#include <stdint.h>

// ---------------------------------------------------------------------------
// Bitmask NMS for gfx1250 (wave32).
//   K1 rank_scatter : O(N^2) stable rank-by-count sort via LDS score tiles.
//   K2 mask_kernel  : suppression bit matrix; TDM (tensor_load_to_lds) stages
//                     the 2KB row-box tile into LDS, wave32 ballot = 32 cols.
//   K3 scan_kernel  : single-wave barrier-free greedy scan; remv bitset lives
//                     in registers (16 words/lane), v_readlane broadcast,
//                     coalesced b32 mask-row loads out of L2 + prefetch.
//   K4 output_kernel: masked writes of sorted boxes/scores + keep-as-float.
// ---------------------------------------------------------------------------

#define NMS_THRESH 0.5f
#define MAX_PROPOSALS 1000
#define MAXW   512          // supports N <= 16384 (N = 10000 here)
#define KSLOTS 16           // 16 slots * 32 lanes = 512 mask words max

typedef unsigned int uint32x4 __attribute__((ext_vector_type(4)));
typedef int          int32x4  __attribute__((ext_vector_type(4)));
typedef int          int32x8  __attribute__((ext_vector_type(8)));

#if __has_builtin(__builtin_amdgcn_tensor_load_to_lds)
#define HAVE_TDM 1
#else
#define HAVE_TDM 0
#endif

__device__ __forceinline__ unsigned ballot32(bool p) {
#if __has_builtin(__builtin_amdgcn_ballot_w32)
  return __builtin_amdgcn_ballot_w32(p);
#else
  return (unsigned)__ballot(p);   // wave32: low 32 bits
#endif
}

__device__ __forceinline__ void wait_tensorcnt0() {
#if __has_builtin(__builtin_amdgcn_s_wait_tensorcnt)
  __builtin_amdgcn_s_wait_tensorcnt(0);
#else
  asm volatile("s_wait_tensorcnt 0x0" ::: "memory");
#endif
}

#if HAVE_TDM
// 1-D TDM copy: ndw dwords from global gptr -> LDS byte offset lds_off.
// D# per CDNA5 ISA: group0 = {flags, lds_addr, gaddr},
// group1 = {data_size=4B, tensor_dim0=tile_dim0=ndw, tensor_dim1=tile_dim1=1}.
// Tile == tensor, so no OOB clamping. Groups 2/3 zero (<=2D tensor).
__device__ __forceinline__ void tdm_load_1d(unsigned lds_off, const void* gptr, int ndw) {
  const unsigned long long ga = (unsigned long long)(uintptr_t)gptr;
  const unsigned T = (unsigned)ndw;
  uint32x4 g0;
  g0[0] = 1u;                                   // count=1, is_restore=0, gather=0
  g0[1] = lds_off;                              // lds_addr (bytes)
  g0[2] = (unsigned)ga;                         // global_addr[31:0]
  g0[3] = (unsigned)((ga >> 32) & 0x1FFFFFFu)   // global_addr[56:32]
        | (2u << 30);                           // type=2 ("image")
  int32x8 g1;
  g1[0] = (int)(2u << 16);                      // workgroup_mask=0, data_size=4B
  g1[1] = (int)((T & 0xFFFFu) << 16);           // atomic_barrier_addr=0, dim0[15:0]
  g1[2] = (int)((T >> 16) | (1u << 16));        // dim0[31:16], tensor_dim1=1
  g1[3] = (int)(T << 16);                       // tensor_dim1 hi=0, tile_dim0=T
  g1[4] = 1;                                    // tile_dim1=1, tile_dim2=0
  g1[5] = (int)T;                               // tensor_dim0_stride lo
  g1[6] = 0;
  g1[7] = 0;
  int32x4 gz = (int32x4)(0);
#if defined(__clang_major__) && (__clang_major__ >= 23)
  int32x8 gz8 = (int32x8)(0);
  __builtin_amdgcn_tensor_load_to_lds(g0, g1, gz, gz, gz8, 0);
#else
  __builtin_amdgcn_tensor_load_to_lds(g0, g1, gz, gz, 0);
#endif
}
#endif

// ---- K1: stable descending rank by counting, then scatter -----------------
__global__ void rank_scatter_kernel(const float* __restrict__ boxes,
                                    const float* __restrict__ scores,
                                    float* __restrict__ sboxes,
                                    float* __restrict__ sscores,
                                    int N) {
  __shared__ float tile[256];
  const int i = blockIdx.x * 256 + threadIdx.x;
  const float si = (i < N) ? scores[i] : 0.0f;
  int cnt = 0;
  for (int base = 0; base < N; base += 256) {
    const int j = base + threadIdx.x;
    tile[threadIdx.x] = (j < N) ? scores[j] : -1.0f;  // scores are in (0,1)
    __syncthreads();
    const int lim = min(256, N - base);
    for (int k = 0; k < lim; ++k) {
      const float sj = tile[k];
      const int jj = base + k;
      // position in argsort(-scores) with stable tie-break by original index
      cnt += (int)((sj > si) || ((sj == si) && (jj < i)));
    }
    __syncthreads();
  }
  if (i < N) {
    sscores[cnt] = si;
    ((float4*)sboxes)[cnt] = ((const float4*)boxes)[i];
  }
}

// ---- K2: suppression bitmask; TDM-staged row tile, wave32 ballot ----------
__global__ void mask_kernel(const float4* __restrict__ sboxes,
                            unsigned* __restrict__ mask,
                            int N, int nWords) {
  __shared__ float4 rowTile[128];               // 2KB row-box tile
  const int lane = threadIdx.x & 31;
  const int wave = threadIdx.x >> 5;
  const int col  = blockIdx.x * 32 + lane;
  const int blockRowBase = blockIdx.y * 128;
  const int rowsHere = min(128, N - blockRowBase);

#if HAVE_TDM
  if (wave == 0) {                               // TDM issues once (EXEC-ignored)
    tdm_load_1d((unsigned)(uintptr_t)&rowTile[0],
                (const void*)(sboxes + blockRowBase), rowsHere * 4);
    wait_tensorcnt0();
  }
  __syncthreads();
#else
  for (int t = threadIdx.x; t < rowsHere; t += 256)
    rowTile[t] = sboxes[blockRowBase + t];
  __syncthreads();
#endif

  float4 cb = make_float4(0.f, 0.f, 0.f, 0.f);
  if (col < N) cb = sboxes[col];                 // global_load_b128, stays in VGPRs
  const float areaC = (cb.z - cb.x) * (cb.w - cb.y);

  const int rowBase = blockRowBase + wave * 16;  // 8 waves * 16 rows per block
  for (int rr = 0; rr < 16; ++rr) {
    const int r = rowBase + rr;
    if (r >= N) break;
    const float4 rb = rowTile[wave * 16 + rr];   // ds_load_b128 broadcast
    const float areaR = (rb.z - rb.x) * (rb.w - rb.y);
    const float ix1 = fmaxf(rb.x, cb.x);
    const float iy1 = fmaxf(rb.y, cb.y);
    const float ix2 = fminf(rb.z, cb.z);
    const float iy2 = fminf(rb.w, cb.w);
    const float iw = fmaxf(ix2 - ix1, 0.0f);
    const float ih = fmaxf(iy2 - iy1, 0.0f);
    const float inter = iw * ih;
    const float uni = areaR + areaC - inter;
    const float iou = inter / fmaxf(uni, 1e-9f);
    const bool pred = (iou > NMS_THRESH) && (col > r) && (col < N);
    const unsigned m = ballot32(pred);
    if (lane == 0) mask[(size_t)r * nWords + blockIdx.x] = m;
  }
}

// ---- K3: single-wave barrier-free greedy scan + cap -----------------------
__global__ __launch_bounds__(32) void scan_kernel(const unsigned* __restrict__ mask,
                                                  unsigned* __restrict__ keepWords,
                                                  int N, int nWords) {
  const int lane = (int)threadIdx.x;             // 0..31, one wave32
  unsigned remv[KSLOTS];                         // lane owns words lane+32k
  unsigned keep[KSLOTS];
#pragma unroll
  for (int k = 0; k < KSLOTS; ++k) { remv[k] = 0u; keep[k] = 0u; }

#pragma unroll
  for (int s = 0; s < KSLOTS; ++s) {             // slot s covers rows [s*1024, ...)
    const int i0 = s << 10;
    if (i0 < N) {
      const int i1 = min(N, i0 + 1024);
      for (int i = i0; i < i1; ++i) {
        const int srcLane = (i >> 5) & 31;       // uniform -> v_readlane to SGPR
        const unsigned rw = (unsigned)__builtin_amdgcn_readlane((int)remv[s], srcLane);
        if (((rw >> (i & 31)) & 1u) == 0u) {     // row i survives: suppress later
          if (lane == srcLane) keep[s] |= (1u << (i & 31));
          const unsigned* row = mask + (size_t)i * nWords;
#pragma unroll
          for (int k = 0; k < KSLOTS; ++k) {
            const int w = lane + (k << 5);
            if (w < nWords) {
              remv[k] |= row[w];                 // coalesced b32, mask is L2-resident
              if (i + 1 < N) __builtin_prefetch(row + nWords + w, 0, 0);
            }
          }
        }
      }
    }
  }

  // cap at MAX_PROPOSALS: stage keep words in LDS, exclusive prefix, trim
  __shared__ unsigned kw[MAXW];
  __shared__ unsigned pre[MAXW];
#pragma unroll
  for (int k = 0; k < KSLOTS; ++k) {
    const int w = lane + (k << 5);
    if (w < nWords) kw[w] = keep[k];
  }
  __syncthreads();                               // single-wave: compiler fence
  if (lane == 0) {
    unsigned run = 0;
    for (int w = 0; w < nWords; ++w) { pre[w] = run; run += (unsigned)__popc(kw[w]); }
  }
  __syncthreads();
  for (int w = lane; w < nWords; w += 32) {
    const unsigned word = kw[w];
    unsigned rank = pre[w];
    unsigned nw = 0u;
    for (int b = 0; b < 32; ++b) {
      if ((word >> b) & 1u) {
        if (rank < (unsigned)MAX_PROPOSALS) nw |= (1u << b);
        ++rank;
      }
    }
    keepWords[w] = nw;
  }
}

// ---- K4: masked outputs ---------------------------------------------------
__global__ void output_kernel(const float4* __restrict__ sboxes,
                              const float* __restrict__ sscores,
                              const unsigned* __restrict__ keepWords,
                              float* __restrict__ out,
                              int N) {
  const int i = blockIdx.x * 256 + threadIdx.x;
  if (i >= N) return;
  const unsigned k = (keepWords[i >> 5] >> (i & 31)) & 1u;
  const float4 b = sboxes[i];
  const float4 z = make_float4(0.f, 0.f, 0.f, 0.f);
  ((float4*)out)[i]   = k ? b : z;               // boxes_out: [0, 4N)
  out[4 * N + i]      = k ? sscores[i] : 0.f;    // scores_out: [4N, 5N)
  out[5 * N + i]      = k ? 1.0f : 0.0f;         // keep:       [5N, 6N)
}

// ---------------------------------------------------------------------------
extern "C" void kernel_launch(void* const* d_in, const int* in_sizes, int n_in,
                              void* d_out, int out_size, void* d_ws, size_t ws_size,
                              hipStream_t stream) {
  const float* boxes  = (const float*)d_in[0];   // (N,4) fp32
  const float* scores = (const float*)d_in[1];   // (N,)  fp32
  const int N = in_sizes[1];
  const int nWords = (N + 31) / 32;

  // workspace layout (N=10000 -> ~12.72 MB total)
  char* ws = (char*)d_ws;
  float*    sboxes    = (float*)ws;                                   // N*4 f32
  float*    sscores   = (float*)(ws + (size_t)N * 16);                // N   f32
  unsigned* keepWords = (unsigned*)(ws + (size_t)N * 16 + (size_t)N * 4);
  size_t maskOff = (size_t)N * 16 + (size_t)N * 4 + (size_t)nWords * 4;
  maskOff = (maskOff + 255) & ~(size_t)255;
  unsigned* mask = (unsigned*)(ws + maskOff);                         // N*nWords u32

  rank_scatter_kernel<<<(N + 255) / 256, 256, 0, stream>>>(boxes, scores, sboxes, sscores, N);

  dim3 gMask(nWords, (N + 127) / 128);
  mask_kernel<<<gMask, 256, 0, stream>>>((const float4*)sboxes, mask, N, nWords);

  scan_kernel<<<1, 32, 0, stream>>>(mask, keepWords, N, nWords);

  output_kernel<<<(N + 255) / 256, 256, 0, stream>>>((const float4*)sboxes, sscores,
                                                     keepWords, (float*)d_out, N);
}